// VocabGraphConvolution_21320217658044
// MI455X (gfx1250) — compile-verified
//
#include <hip/hip_runtime.h>
#include <hip/hip_bf16.h>
#include <stdint.h>

// Problem constants (match reference)
#define BB    8
#define SS    512
#define EE    768
#define VV    30522
#define GVV   20000
#define NNZV  400000
#define HH    64
#define OO    16

#define WPB     640                 // bitmap words per batch (625 used, padded)
#define NCHUNK  8
#define CHUNK   (GVV / NCHUNK)      // 2500
#define KSTEPS  (CHUNK / 4)         // 625
#define ETILES  (EE / 16)           // 48 (each wave covers 16 E-rows now)

typedef float v2f __attribute__((ext_vector_type(2)));
typedef float v8f __attribute__((ext_vector_type(8)));

// ---------------- workspace layout ----------------
// [0, 20480)                       : present bitmap (8 * 640 u32)
// [20480, +B*GV*H*4)               : H_vh  (8, 20000, 64) f32  = 40.96 MB
// [.., +NCHUNK*B*E*H*4)            : chunk partials of fused_H = 12.58 MB
#define PRESENT_BYTES  20480
#define HVH_BYTES      ((size_t)BB * GVV * HH * 4)
#define PART_BYTES     ((size_t)NCHUNK * BB * EE * HH * 4)

__global__ void gcn_zero(uint32_t* __restrict__ p, long n) {
  long i = (long)blockIdx.x * blockDim.x + threadIdx.x;
  long stride = (long)gridDim.x * blockDim.x;
  for (; i < n; i += stride) p[i] = 0u;
}

__global__ void gcn_present(const int* __restrict__ input_ids,
                            const int* __restrict__ tok2g,
                            uint32_t* __restrict__ present) {
  int tid = blockIdx.x * blockDim.x + threadIdx.x;
  if (tid >= BB * SS) return;
  int b = tid / SS;
  int id = input_ids[tid];
  if (id > VV - 1 || id < 0) id = 0;         // reference: OOR ids -> 0
  int g = tok2g[id];
  atomicOr(&present[b * WPB + (g >> 5)], 1u << (g & 31));
}

// 64 threads (2 waves) per nnz entry: thread h handles one H column.
__global__ void gcn_scatter(const int* __restrict__ rows,
                            const int* __restrict__ cols,
                            const float* __restrict__ vals,
                            const float* __restrict__ W,
                            const uint32_t* __restrict__ present,
                            float* __restrict__ Hvh) {
  int k = blockIdx.x * 4 + (threadIdx.x >> 6);
  const int h = threadIdx.x & 63;
  const int kStride = gridDim.x * 4;
  for (; k < NNZV; k += kStride) {
    const int row = rows[k];
    const int col = cols[k];
    const float vw = vals[k] * W[(size_t)col * HH + h];
    const int rw = row >> 5, rbit = row & 31;
    const int cw = col >> 5, cbit = col & 31;
#pragma unroll
    for (int b = 0; b < BB; ++b) {
      const uint32_t m = (present[b * WPB + rw] >> rbit) |
                         (present[b * WPB + cw] >> cbit);
      if (m & 1u)
        atomicAdd(&Hvh[((size_t)b * GVV + row) * HH + h], vw);
    }
  }
}

// fused_H partial GEMM: per wave, 16(E) x 64(H) tile over one 2500-wide K chunk.
// A (16x4 f32)  : A[m,k] = emb[gv_ids[v0+k]][e0+m]   (gvocab_ev^T tile)
// B (4x16 f32)  : B[k,n] = Hvh[b][v0+k][ht*16+n]
// f32 VGPR layout (ISA 7.12.2): A lane<16 -> M=lane, vgpr0=K0 vgpr1=K1;
// lanes 16..31 -> K2/K3.  C/D: vgpr r -> M=r (lanes<16) / M=r+8 (lanes>=16).
// 4 x v8f accumulators = 32 VGPRs -> no spills (the 2-Mtile version spilled).
__global__ void __launch_bounds__(128)
gcn_gemm_wmma(const float* __restrict__ emb,
              const int* __restrict__ gv_ids,
              const float* __restrict__ Hvh,
              float* __restrict__ part) {
  const int wave   = blockIdx.x * 4 + (threadIdx.x >> 5);
  const int lane   = threadIdx.x & 31;
  const int lane16 = lane & 15;
  const int hi     = lane >> 4;

  const int c  = wave % NCHUNK;
  const int et = (wave / NCHUNK) % ETILES;
  const int b  = wave / (NCHUNK * ETILES);

  const int e0    = et * 16;
  const int vbase = c * CHUNK;
  const float* __restrict__ HvhB = Hvh + (size_t)b * GVV * HH;

  const v8f zacc = {0.f, 0.f, 0.f, 0.f, 0.f, 0.f, 0.f, 0.f};
  v8f acc0 = zacc, acc1 = zacc, acc2 = zacc, acc3 = zacc;

  for (int step = 0; step < KSTEPS; ++step) {
    const int va = vbase + step * 4 + 2 * hi;   // this half-wave's K pair
    const int g0 = gv_ids[va];
    const int g1 = gv_ids[va + 1];

    v2f a;
    a.x = emb[(size_t)g0 * EE + e0 + lane16];
    a.y = emb[(size_t)g1 * EE + e0 + lane16];

    const float* __restrict__ r0 = HvhB + (size_t)va * HH + lane16;
    const float* __restrict__ r1 = r0 + HH;

    v2f b0, b1, b2, b3;
    b0.x = r0[0];  b0.y = r1[0];
    b1.x = r0[16]; b1.y = r1[16];
    b2.x = r0[32]; b2.y = r1[32];
    b3.x = r0[48]; b3.y = r1[48];

    acc0 = __builtin_amdgcn_wmma_f32_16x16x4_f32(false, a, false, b0,
                                                 (short)0, acc0, false, false);
    acc1 = __builtin_amdgcn_wmma_f32_16x16x4_f32(false, a, false, b1,
                                                 (short)0, acc1, false, false);
    acc2 = __builtin_amdgcn_wmma_f32_16x16x4_f32(false, a, false, b2,
                                                 (short)0, acc2, false, false);
    acc3 = __builtin_amdgcn_wmma_f32_16x16x4_f32(false, a, false, b3,
                                                 (short)0, acc3, false, false);
  }

  // store this chunk's partial tile (exclusive per wave -> plain stores)
  float* __restrict__ pb =
      part + (((size_t)c * BB + b) * EE + e0) * HH + lane16;
  v8f accs[4] = {acc0, acc1, acc2, acc3};
#pragma unroll
  for (int t = 0; t < 4; ++t)
#pragma unroll
    for (int r = 0; r < 8; ++r) {
      const int e = r + 8 * hi;            // row within the 16-row tile
      pb[(size_t)e * HH + t * 16] = accs[t][r];
    }
}

// reduce NCHUNK partials and apply fc: out[b,o,e] = sum_h fused[b,e,h]*fc_w[o,h] + fc_b[o]
__global__ void gcn_final(const float* __restrict__ part,
                          const float* __restrict__ fc_w,
                          const float* __restrict__ fc_b,
                          float* __restrict__ out) {
  __shared__ float lds[HH];
  const int e = blockIdx.x % EE;
  const int b = blockIdx.x / EE;
  const int h = threadIdx.x;
  float f = 0.f;
#pragma unroll
  for (int c = 0; c < NCHUNK; ++c)
    f += part[(((size_t)c * BB + b) * EE + e) * HH + h];
  lds[h] = f;
  __syncthreads();
  if (h < OO) {
    float a = fc_b[h];
#pragma unroll
    for (int hh = 0; hh < HH; ++hh) a += lds[hh] * fc_w[h * HH + hh];
    out[((size_t)b * OO + h) * EE + e] = a;
  }
}

extern "C" void kernel_launch(void* const* d_in, const int* in_sizes, int n_in,
                              void* d_out, int out_size, void* d_ws, size_t ws_size,
                              hipStream_t stream) {
  (void)in_sizes; (void)n_in; (void)out_size;
  const float* emb        = (const float*)d_in[0];
  const float* adj_values = (const float*)d_in[1];
  const float* W_vh       = (const float*)d_in[2];
  const float* fc_w       = (const float*)d_in[3];
  const float* fc_b       = (const float*)d_in[4];
  const int*   input_ids  = (const int*)d_in[5];
  const int*   tok2g      = (const int*)d_in[6];
  const int*   gv_ids     = (const int*)d_in[7];
  const int*   adj_rows   = (const int*)d_in[8];
  const int*   adj_cols   = (const int*)d_in[9];
  float* out = (float*)d_out;

  const size_t need = PRESENT_BYTES + HVH_BYTES + PART_BYTES;
  if (ws_size < need) return;  // ~53.6 MB required

  uint8_t*  ws      = (uint8_t*)d_ws;
  uint32_t* present = (uint32_t*)ws;
  float*    Hvh     = (float*)(ws + PRESENT_BYTES);
  float*    part    = (float*)(ws + PRESENT_BYTES + HVH_BYTES);

  // 1) zero present + Hvh + partials (accumulators)
  gcn_zero<<<4096, 256, 0, stream>>>((uint32_t*)ws, (long)(need / 4));
  // 2) membership bitmap
  gcn_present<<<(BB * SS) / 256, 256, 0, stream>>>(input_ids, tok2g, present);
  // 3) masked sparse scatter-add into H_vh
  gcn_scatter<<<8192, 256, 0, stream>>>(adj_rows, adj_cols, adj_values, W_vh,
                                        present, Hvh);
  // 4) WMMA GEMM: 8 b * 48 etiles * 8 chunks = 3072 waves, 4 waves/block
  gcn_gemm_wmma<<<(BB * ETILES * NCHUNK) / 4, 128, 0, stream>>>(emb, gv_ids,
                                                                Hvh, part);
  // 5) chunk reduction + FC epilogue
  gcn_final<<<BB * EE, HH, 0, stream>>>(part, fc_w, fc_b, out);
}